// MyGRU_63539746177495
// MI455X (gfx1250) — compile-verified
//
#include <hip/hip_runtime.h>
#include <hip/hip_bf16.h>
#include <math.h>

// Problem constants (match reference)
#define SEQ   128
#define BAT   64
#define EDIM  512
#define PLEN  160
#define PDIM  512
#define HDIM  1024
#define XDIM  1024   // E + P_DIM
#define NEGBIG -1000000000.0f

typedef float v2f __attribute__((ext_vector_type(2)));
typedef float v8f __attribute__((ext_vector_type(8)));

static __device__ __forceinline__ v8f wmma_f32(v2f a, v2f b, v8f c) {
    // D = A(16x4) * B(4x16) + C  in fp32 on the matrix pipe
    return __builtin_amdgcn_wmma_f32_16x16x4_f32(false, a, false, b, (short)0, c, false, false);
}

// ---------------------------------------------------------------------------
// h0 broadcast: h0[b, j] = h_init[j]
// ---------------------------------------------------------------------------
__global__ void k_init_h(const float* __restrict__ h_init, float* __restrict__ h0) {
    int idx = blockIdx.x * blockDim.x + threadIdx.x;   // 0 .. B*H-1
    h0[idx] = h_init[idx & (HDIM - 1)];
}

// ---------------------------------------------------------------------------
// query = h @ wq^T + bq     (M=64, N=512, K=1024), WMMA fp32
// grid: 16 WGs x 256 threads; WG owns cols [g*32, g*32+32).
// Each wave owns one 16x16 output position; K is split into two interleaved
// streams (two accumulators) for 2 independent WMMA chains per wave.
// ---------------------------------------------------------------------------
__global__ void k_query(const float* __restrict__ h,      // [B, H]
                        const float* __restrict__ wq,     // [PDIM, H]
                        const float* __restrict__ bq,     // [PDIM]
                        float* __restrict__ query) {      // [B, PDIM]
    const int j0    = blockIdx.x * 32;
    const int wave  = threadIdx.x >> 5;
    const int lane  = threadIdx.x & 31;
    const int lhalf = (lane >= 16) ? 1 : 0;
    const int l16   = lane & 15;
    const int koff  = lhalf * 2;

    const int mi = wave & 3;                   // 4 row blocks of 16
    const int ni = wave >> 2;                  // 2 col blocks of 16
    const int mrow = mi * 16 + l16;            // A-fragment row (batch)
    const int jq   = j0 + ni * 16 + l16;       // B-fragment col (PDIM)

    const float* __restrict__ arow = h  + (size_t)mrow * HDIM;
    const float* __restrict__ brow = wq + (size_t)jq   * HDIM; // wq[jq, k]

    const v8f vzero = {0.f,0.f,0.f,0.f,0.f,0.f,0.f,0.f};
    v8f acc0 = vzero, acc1 = vzero;

    for (int k = 0; k < HDIM; k += 8) {
        const int ka = k + koff;
        const int kb = k + 4 + koff;
        v2f a0 = *(const v2f*)(arow + ka);
        v2f b0 = *(const v2f*)(brow + ka);
        v2f a1 = *(const v2f*)(arow + kb);
        v2f b1 = *(const v2f*)(brow + kb);
        acc0 = wmma_f32(a0, b0, acc0);
        acc1 = wmma_f32(a1, b1, acc1);
    }
    const v8f acc = acc0 + acc1;

    const float bias = bq[jq];
    #pragma unroll
    for (int e = 0; e < 8; ++e) {
        int m = mi * 16 + e + lhalf * 8;       // C/D row mapping
        query[(size_t)m * PDIM + jq] = acc[e] + bias;
    }
}

// ---------------------------------------------------------------------------
// Attention: per batch element b (one WG each):
//   scores[p] = post[p,b,:] . query[b,:]  (masked)  -> softmax over p
//   context[b,:] = sum_p attn[p] * post[p,b,:]
// grid: 64 WGs x 256 threads (8 waves). global_prefetch_b8 covers the next
// 2KB post row per wave while the current dot-product is in flight.
// ---------------------------------------------------------------------------
__global__ void k_attention(const float* __restrict__ post,     // [P, B, PDIM]
                            const float* __restrict__ query,    // [B, PDIM]
                            const int*   __restrict__ post_len, // [B]
                            float* __restrict__ context) {      // [B, PDIM]
    __shared__ float s_q[PDIM];
    __shared__ float s_sc[PLEN];

    const int b    = blockIdx.x;
    const int tid  = threadIdx.x;
    const int wave = tid >> 5;
    const int lane = tid & 31;
    const int plen = post_len[b];

    // stage query row
    s_q[tid]       = query[(size_t)b * PDIM + tid];
    s_q[tid + 256] = query[(size_t)b * PDIM + tid + 256];
    __syncthreads();

    // scores: each wave handles p = wave, wave+8, ...
    for (int p = wave; p < PLEN; p += 8) {
        const float* __restrict__ prow = post + ((size_t)p * BAT + b) * PDIM;
        if (p + 8 < PLEN) {
            // prefetch next row (2KB) at 64B lane stride -> global_prefetch_b8
            const char* nxt = (const char*)(post + ((size_t)(p + 8) * BAT + b) * PDIM);
            __builtin_prefetch(nxt + lane * 64, 0, 3);
        }
        float s = 0.f;
        for (int d = lane; d < PDIM; d += 32) s += prow[d] * s_q[d];
        #pragma unroll
        for (int off = 16; off >= 1; off >>= 1) s += __shfl_xor(s, off, 32);
        if (lane == 0) s_sc[p] = (p < plen) ? s : NEGBIG;
    }
    __syncthreads();

    // masked softmax over P (wave 0, butterfly reductions)
    if (wave == 0) {
        float mx = -3.4e38f;
        for (int p = lane; p < PLEN; p += 32) mx = fmaxf(mx, s_sc[p]);
        #pragma unroll
        for (int off = 16; off >= 1; off >>= 1) mx = fmaxf(mx, __shfl_xor(mx, off, 32));
        float sum = 0.f;
        float e0[(PLEN + 31) / 32];
        int c = 0;
        for (int p = lane; p < PLEN; p += 32, ++c) {
            float e = __expf(s_sc[p] - mx);
            e0[c] = e;
            sum += e;
        }
        #pragma unroll
        for (int off = 16; off >= 1; off >>= 1) sum += __shfl_xor(sum, off, 32);
        const float inv = 1.f / sum;
        c = 0;
        for (int p = lane; p < PLEN; p += 32, ++c) s_sc[p] = e0[c] * inv;
    }
    __syncthreads();

    // context: each thread owns 2 adjacent dims
    const int d = tid * 2;
    float a0 = 0.f, a1 = 0.f;
    for (int p = 0; p < PLEN; ++p) {
        const float a = s_sc[p];
        const v2f pv = *(const v2f*)(post + ((size_t)p * BAT + b) * PDIM + d);
        a0 += a * pv.x;
        a1 += a * pv.y;
    }
    context[(size_t)b * PDIM + d]     = a0;
    context[(size_t)b * PDIM + d + 1] = a1;
}

// ---------------------------------------------------------------------------
// Fused GRU cell: gi = [xt|ctx] @ W_ih^T + b_ih ; gh = h @ W_hh^T + b_hh
//   r = sig(i_r+h_r); z = sig(i_z+h_z); n = tanh(i_n + r*h_n)
//   h_new = (1-z)*n + z*h ; masked by (length > t)
// grid: 32 WGs x 256 threads; WG g owns H-cols [g*32, g*32+32) across all 3
// gate segments (cols j, j+H, j+2H) so the elementwise math fuses locally.
// Each wave owns one 16x16 output position with 6 independent fp32 WMMA
// accumulator chains (r/z/n x ih/hh) -> deep ILP, no barriers in the K loop.
// ---------------------------------------------------------------------------
__global__ void k_gru_gates(const float* __restrict__ xt,     // [B, E] = incoming[t]
                            const float* __restrict__ ctx,    // [B, PDIM]
                            const float* __restrict__ hprev,  // [B, H]
                            const float* __restrict__ wih,    // [3H, XDIM]
                            const float* __restrict__ whh,    // [3H, H]
                            const float* __restrict__ bih,    // [3H]
                            const float* __restrict__ bhh,    // [3H]
                            const int*   __restrict__ length, // [B]
                            const int t,
                            float* __restrict__ hout) {       // [B, H] = hs[t]
    const int j0    = blockIdx.x * 32;
    const int wave  = threadIdx.x >> 5;
    const int lane  = threadIdx.x & 31;
    const int lhalf = (lane >= 16) ? 1 : 0;
    const int l16   = lane & 15;
    const int koff  = lhalf * 2;

    const int mi = wave & 3;                 // 4 row blocks of 16 (batch)
    const int ni = wave >> 2;                // 2 col blocks of 16
    const int mrow = mi * 16 + l16;
    const int jr   = j0 + ni * 16 + l16;     // col in [0,H) (r-gate col)

    const float* __restrict__ xrow_in = xt    + (size_t)mrow * EDIM;
    const float* __restrict__ xrow_cx = ctx   + (size_t)mrow * PDIM;
    const float* __restrict__ hrow    = hprev + (size_t)mrow * HDIM;

    const float* __restrict__ wih_r = wih + (size_t)(jr)            * XDIM;
    const float* __restrict__ wih_z = wih + (size_t)(jr + HDIM)     * XDIM;
    const float* __restrict__ wih_n = wih + (size_t)(jr + 2 * HDIM) * XDIM;
    const float* __restrict__ whh_r = whh + (size_t)(jr)            * HDIM;
    const float* __restrict__ whh_z = whh + (size_t)(jr + HDIM)     * HDIM;
    const float* __restrict__ whh_n = whh + (size_t)(jr + 2 * HDIM) * HDIM;

    const v8f vzero = {0.f,0.f,0.f,0.f,0.f,0.f,0.f,0.f};
    v8f acc[6];
    #pragma unroll
    for (int i = 0; i < 6; ++i) acc[i] = vzero;

    for (int k = 0; k < XDIM; k += 4) {
        const int kk = k + koff;
        // x = [incoming | context]; the 512 split is 4-aligned => uniform select
        const float* xr = (kk < EDIM) ? (xrow_in + kk) : (xrow_cx + (kk - EDIM));
        const v2f ax = *(const v2f*)xr;
        const v2f ah = *(const v2f*)(hrow + kk);
        const v2f br_i = *(const v2f*)(wih_r + kk);
        const v2f bz_i = *(const v2f*)(wih_z + kk);
        const v2f bn_i = *(const v2f*)(wih_n + kk);
        const v2f br_h = *(const v2f*)(whh_r + kk);
        const v2f bz_h = *(const v2f*)(whh_z + kk);
        const v2f bn_h = *(const v2f*)(whh_n + kk);
        acc[0] = wmma_f32(ax, br_i, acc[0]);
        acc[1] = wmma_f32(ax, bz_i, acc[1]);
        acc[2] = wmma_f32(ax, bn_i, acc[2]);
        acc[3] = wmma_f32(ah, br_h, acc[3]);
        acc[4] = wmma_f32(ah, bz_h, acc[4]);
        acc[5] = wmma_f32(ah, bn_h, acc[5]);
    }

    const float b_ir = bih[jr], b_iz = bih[jr + HDIM], b_in = bih[jr + 2 * HDIM];
    const float b_hr = bhh[jr], b_hz = bhh[jr + HDIM], b_hn = bhh[jr + 2 * HDIM];

    #pragma unroll
    for (int e = 0; e < 8; ++e) {
        const int m = mi * 16 + e + lhalf * 8;     // C/D row mapping
        const float ir = acc[0][e] + b_ir;
        const float iz = acc[1][e] + b_iz;
        const float in_ = acc[2][e] + b_in;
        const float hr = acc[3][e] + b_hr;
        const float hz = acc[4][e] + b_hz;
        const float hn = acc[5][e] + b_hn;
        const float r = 1.f / (1.f + __expf(-(ir + hr)));
        const float z = 1.f / (1.f + __expf(-(iz + hz)));
        const float n = tanhf(in_ + r * hn);
        const float hold = hprev[(size_t)m * HDIM + jr];
        float hnew = (1.f - z) * n + z * hold;
        hnew = (length[m] > t) ? hnew : 0.f;
        hout[(size_t)m * HDIM + jr] = hnew;
    }
}

// ---------------------------------------------------------------------------
extern "C" void kernel_launch(void* const* d_in, const int* in_sizes, int n_in,
                              void* d_out, int out_size, void* d_ws, size_t ws_size,
                              hipStream_t stream) {
    (void)in_sizes; (void)n_in; (void)out_size; (void)ws_size;

    const float* incoming  = (const float*)d_in[0];   // [S, B, E]
    const float* post      = (const float*)d_in[1];   // [P, B, PDIM]
    const float* h_init    = (const float*)d_in[2];   // [1, 1, H]
    const float* weight_ih = (const float*)d_in[3];   // [3H, XDIM]
    const float* weight_hh = (const float*)d_in[4];   // [3H, H]
    const float* bias_ih   = (const float*)d_in[5];   // [3H]
    const float* bias_hh   = (const float*)d_in[6];   // [3H]
    const float* wq        = (const float*)d_in[7];   // [PDIM, H]
    const float* bq        = (const float*)d_in[8];   // [PDIM]
    const int*   length    = (const int*)d_in[9];     // [B]
    const int*   post_len  = (const int*)d_in[10];    // [B]

    float* out = (float*)d_out;                       // hs [S,B,H] ++ h_last [B,H]

    float* ws      = (float*)d_ws;
    float* h0      = ws;                              // [B, H]
    float* query   = h0 + BAT * HDIM;                 // [B, PDIM]
    float* context = query + BAT * PDIM;              // [B, PDIM]

    // h0 = broadcast(h_init)
    k_init_h<<<(BAT * HDIM) / 256, 256, 0, stream>>>(h_init, h0);

    for (int t = 0; t < SEQ; ++t) {
        const float* hcur = (t == 0) ? h0 : (out + (size_t)(t - 1) * BAT * HDIM);
        float* hnext = out + (size_t)t * BAT * HDIM;

        k_query<<<PDIM / 32, 256, 0, stream>>>(hcur, wq, bq, query);
        k_attention<<<BAT, 256, 0, stream>>>(post, query, post_len, context);
        k_gru_gates<<<HDIM / 32, 256, 0, stream>>>(incoming + (size_t)t * BAT * EDIM,
                                                   context, hcur,
                                                   weight_ih, weight_hh,
                                                   bias_ih, bias_hh,
                                                   length, t, hnext);
    }

    // h_last = hs[S-1]
    hipMemcpyAsync(out + (size_t)SEQ * BAT * HDIM,
                   out + (size_t)(SEQ - 1) * BAT * HDIM,
                   (size_t)BAT * HDIM * sizeof(float),
                   hipMemcpyDeviceToDevice, stream);
}